// ECELoss_84628035600455
// MI455X (gfx1250) — compile-verified
//
#include <hip/hip_runtime.h>

typedef float v4f __attribute__((ext_vector_type(4)));
typedef float v2f __attribute__((ext_vector_type(2)));
typedef float v8f __attribute__((ext_vector_type(8)));

#define N_BINS 20
#define BLOCK 256
#define WAVES_PER_BLOCK 8   // 256 threads / wave32

// ---------------------------------------------------------------------------
// Pass 1: one wave32 per row (32 lanes x float4 = 128 classes).
// Streaming, non-temporal b128 loads; wave-level max/argmax + sum(exp);
// per-wave LDS bin accumulators (no atomics -> bitwise deterministic);
// per-block partial histogram written to d_ws.
// ---------------------------------------------------------------------------
__global__ __launch_bounds__(BLOCK) void ece_rows(const float* __restrict__ logits,
                                                  const int*   __restrict__ labels,
                                                  float*       __restrict__ partials,
                                                  int n)
{
    __shared__ float sbins[WAVES_PER_BLOCK][2 * N_BINS];
    const int lane = threadIdx.x & 31;
    const int wav  = threadIdx.x >> 5;

    for (int i = threadIdx.x; i < WAVES_PER_BLOCK * 2 * N_BINS; i += BLOCK)
        (&sbins[0][0])[i] = 0.0f;
    __syncthreads();

    const int gw     = blockIdx.x * WAVES_PER_BLOCK + wav;
    const int stride = gridDim.x * WAVES_PER_BLOCK;

    for (int row = gw; row < n; row += stride) {
        const v4f* p = (const v4f*)(logits + (size_t)row * 128);
        v4f v = __builtin_nontemporal_load(p + lane);   // TH=NT: 512MB stream, read once

        // per-lane max + first-occurrence index
        float m = v.x; int mi = 0;
        if (v.y > m) { m = v.y; mi = 1; }
        if (v.z > m) { m = v.z; mi = 2; }
        if (v.w > m) { m = v.w; mi = 3; }
        int   wi = (lane << 2) | mi;
        float wm = m;

        // wave max with min-index tie-break (matches jnp.argmax first-occurrence)
        #pragma unroll
        for (int off = 16; off > 0; off >>= 1) {
            float om = __shfl_xor(wm, off, 32);
            int   oi = __shfl_xor(wi, off, 32);
            if (om > wm || (om == wm && oi < wi)) { wm = om; wi = oi; }
        }

        // sum of exp(l - max); the max element contributes exp(0)=1
        float s = __expf(v.x - wm) + __expf(v.y - wm) +
                  __expf(v.z - wm) + __expf(v.w - wm);
        #pragma unroll
        for (int off = 16; off > 0; off >>= 1)
            s += __shfl_xor(s, off, 32);

        if (lane == 0) {
            float conf = 1.0f / s;                 // max softmax prob
            int b = (int)ceilf(conf * (float)N_BINS) - 1;   // searchsorted(side='left')
            b = b < 0 ? 0 : (b > N_BINS - 1 ? N_BINS - 1 : b);
            float acc = (labels[row] == wi) ? 1.0f : 0.0f;
            sbins[wav][b]          += conf;
            sbins[wav][N_BINS + b] += acc;
        }
    }
    __syncthreads();

    if (threadIdx.x < 2 * N_BINS) {
        float t = 0.0f;
        #pragma unroll
        for (int w = 0; w < WAVES_PER_BLOCK; ++w) t += sbins[w][threadIdx.x];
        partials[(size_t)blockIdx.x * (2 * N_BINS) + threadIdx.x] = t;
    }
}

// ---------------------------------------------------------------------------
// Pass 2: column-sum of partials[nblk][40] via V_WMMA_F32_16X16X4_F32.
// A (16x4 f32, documented layout: lane l, vgpr v -> A[l%16][v+2*(l/16)]) holds
// 16 bins x 4 blocks of partials; B is all-ones (layout independent);
// D[m][*] accumulates the exact f32 per-bin sum over all blocks, fixed order.
// ---------------------------------------------------------------------------
__global__ __launch_bounds__(128) void ece_reduce(const float* __restrict__ partials,
                                                  float*       __restrict__ out,
                                                  int nblk, int n)
{
    __shared__ float bins[64];
    const int lane = threadIdx.x & 31;
    const int wav  = threadIdx.x >> 5;

    if (threadIdx.x < 64) bins[threadIdx.x] = 0.0f;
    __syncthreads();

    if (wav < 3) {                                    // 3 tiles of 16 bins cover 40
        const int   col  = wav * 16 + (lane & 15);    // global bin column
        const float msk  = (col < 2 * N_BINS) ? 1.0f : 0.0f;
        const int   colc = (col < 2 * N_BINS) ? col : 0;
        const int   kb   = 2 * (lane >> 4);           // K-slot base for this half-wave

        v2f bmat; bmat.x = 1.0f; bmat.y = 1.0f;       // ones 4x16
        v8f c = {0.0f, 0.0f, 0.0f, 0.0f, 0.0f, 0.0f, 0.0f, 0.0f};

        for (int k = 0; k < nblk; k += 4) {
            const int r0 = k + kb;
            v2f a;
            a.x = partials[(size_t)r0       * (2 * N_BINS) + colc] * msk; // K = kb
            a.y = partials[(size_t)(r0 + 1) * (2 * N_BINS) + colc] * msk; // K = kb+1
            c = __builtin_amdgcn_wmma_f32_16x16x4_f32(
                    false, a, false, bmat, (short)0, c, false, false);
        }

        // D layout: lane l holds N=l%16, M = r + 8*(l/16); take N==0 lanes
        if ((lane & 15) == 0) {
            const int mb = wav * 16 + ((lane >> 4) << 3);
            #pragma unroll
            for (int r = 0; r < 8; ++r) bins[mb + r] = c[r];
        }
    }
    __syncthreads();

    if (threadIdx.x == 0) {
        float e = 0.0f;
        #pragma unroll
        for (int i = 0; i < N_BINS; ++i)
            e += fabsf(bins[i] - bins[N_BINS + i]);
        out[0] = e / (float)n;
    }
}

// ---------------------------------------------------------------------------
extern "C" void kernel_launch(void* const* d_in, const int* in_sizes, int n_in,
                              void* d_out, int out_size, void* d_ws, size_t ws_size,
                              hipStream_t stream)
{
    const float* logits = (const float*)d_in[0];
    const int*   labels = (const int*)  d_in[1];
    float*       out    = (float*)d_out;
    const int    n      = in_sizes[1];          // N rows (label count)

    int nblk = 1024;                            // 8192 waves: ~122 rows/wave
    const size_t per_blk = (size_t)(2 * N_BINS) * sizeof(float);
    if (ws_size < (size_t)nblk * per_blk) {
        nblk = (int)(ws_size / per_blk);
        nblk &= ~3;                              // WMMA consumes blocks 4 at a time
        if (nblk < 4) nblk = 4;
    }
    float* partials = (float*)d_ws;              // fully overwritten each call

    ece_rows  <<<nblk, BLOCK, 0, stream>>>(logits, labels, partials, n);
    ece_reduce<<<1,    128,   0, stream>>>(partials, out, nblk, n);
}